// GCN_13718125543731
// MI455X (gfx1250) — compile-verified
//
#include <hip/hip_runtime.h>
#include <hip/hip_bf16.h>

// GCN mean-aggregation (copy_u + mean reduce) for MI455X / gfx1250.
// feature: [100000, 64] f32, src/dst: [1.25M] int32 (harness dtype), out: [100000, 64] f32.
//
// Strategy: 3 kernels on `stream`:
//   1) zero the sum accumulator (d_out) and degree array (d_ws)      -> global_store_b128
//   2) edge scatter: 16 threads/edge, float4 gather + 4 f32 atomics  -> global_load_b128 + global_atomic_add_f32
//   3) finalize: mean = sum / max(deg,1)                              -> global_load_b128/global_store_b128
// Feature table (25.6MB) and accumulator (25.6MB) are L2-resident (192MB L2);
// atomics resolve at L2. Bottleneck is L2 atomic throughput, so keep atomics
// independent and unserialized across lanes.

#define GCN_N_NODES 100000
#define GCN_FEAT    64
#define GCN_CHUNKS  16   // 16 x float4 per 64-float row

__global__ void gcn_zero_f4(float4* __restrict__ p, int n4) {
    int i = blockIdx.x * blockDim.x + threadIdx.x;
    if (i < n4) {
        p[i] = make_float4(0.f, 0.f, 0.f, 0.f);
    }
}

__global__ void gcn_scatter(const float* __restrict__ feat,
                            const int*   __restrict__ src,
                            const int*   __restrict__ dst,
                            float*       __restrict__ out_sum,
                            float*       __restrict__ deg,
                            int n_edges) {
    int t = blockIdx.x * blockDim.x + threadIdx.x;
    int e = t >> 4;        // edge index (16 threads per edge)
    int c = t & 15;        // float4 chunk within the 64-float row
    if (e >= n_edges) return;

    int s = src[e];        // 16 lanes hit the same address -> single coalesced request
    int d = dst[e];

    // Coalesced 256B row gather: 16 lanes x global_load_b128
    const float4* frow = (const float4*)(feat + (size_t)s * GCN_FEAT);
    float4 v = frow[c];

    // Fire-and-forget f32 atomics (no-return -> STOREcnt, resolved at L2)
    float* orow = out_sum + (size_t)d * GCN_FEAT + (size_t)c * 4;
    atomicAdd(orow + 0, v.x);
    atomicAdd(orow + 1, v.y);
    atomicAdd(orow + 2, v.z);
    atomicAdd(orow + 3, v.w);

    if (c == 0) {
        atomicAdd(deg + d, 1.0f);
    }
}

__global__ void gcn_finalize(float* __restrict__ out_sum,
                             const float* __restrict__ deg) {
    int t = blockIdx.x * blockDim.x + threadIdx.x;
    int n = t >> 4;
    int c = t & 15;
    if (n >= GCN_N_NODES) return;

    float dg  = deg[n];
    float den = (dg > 0.f) ? dg : 1.0f;   // deg==0 rows hold exact 0 -> 0/1 = 0

    float4* orow = (float4*)(out_sum + (size_t)n * GCN_FEAT) + c;
    float4 s = *orow;
    s.x = s.x / den;
    s.y = s.y / den;
    s.z = s.z / den;
    s.w = s.w / den;
    *orow = s;
}

extern "C" void kernel_launch(void* const* d_in, const int* in_sizes, int n_in,
                              void* d_out, int out_size, void* d_ws, size_t ws_size,
                              hipStream_t stream) {
    const float* feat = (const float*)d_in[0];
    const int*   src  = (const int*)d_in[1];
    const int*   dst  = (const int*)d_in[2];
    const int n_edges = in_sizes[1];

    float* out_sum = (float*)d_out;   // accumulator, then final means (in place)
    float* deg     = (float*)d_ws;    // 100000 floats = 400KB of scratch

    const int threads = 256;          // 8 wave32s per block

    // 1) zero accumulator + degrees (harness poisons buffers; must re-zero every call)
    {
        int out4 = GCN_N_NODES * GCN_FEAT / 4;   // 1,600,000 float4
        gcn_zero_f4<<<(out4 + threads - 1) / threads, threads, 0, stream>>>(
            (float4*)out_sum, out4);
        int deg4 = GCN_N_NODES / 4;              // 25,000 float4
        gcn_zero_f4<<<(deg4 + threads - 1) / threads, threads, 0, stream>>>(
            (float4*)deg, deg4);
    }

    // 2) edge scatter: 16 threads per edge
    {
        long long total  = (long long)n_edges * GCN_CHUNKS;   // 20M threads
        int       blocks = (int)((total + threads - 1) / threads);
        gcn_scatter<<<blocks, threads, 0, stream>>>(feat, src, dst, out_sum, deg, n_edges);
    }

    // 3) finalize: divide by degree in place
    {
        int total = GCN_N_NODES * GCN_CHUNKS;    // 1.6M threads
        gcn_finalize<<<(total + threads - 1) / threads, threads, 0, stream>>>(out_sum, deg);
    }
}